// FSSwishLayer_11776800325729
// MI455X (gfx1250) — compile-verified
//
#include <hip/hip_runtime.h>
#include <stdint.h>

// ---------------------------------------------------------------------------
// FS-Swish spiking-threshold accumulation, MI455X (gfx1250).
//
// Memory-bound: 512 MB streaming traffic -> ~22 us floor at 23.3 TB/s.
// Per element: 16-step serial recurrence at 4 VALU ops/step
// (v_cmp + v_cndmask + 2x v_fma). Four independent chains per thread
// (float4) for VOPD dual-issue. Input staged through a 4-deep CDNA5
// async global->LDS ring (ASYNCcnt), NT stores for the output stream
// (512 MB stream > 192 MB L2, zero reuse). Trip count computed on host
// to avoid the in-kernel 64-bit scalar division blob.
// ---------------------------------------------------------------------------

typedef float v4f __attribute__((ext_vector_type(4)));
// Exact pointee type of the async-copy builtin's pointer params
// (clang prototype: 'int __attribute__((vector_size(16))) __device__ *').
typedef int v4i_vs __attribute__((vector_size(16)));
typedef __attribute__((address_space(1))) v4i_vs* gbl_v4i_p;
typedef __attribute__((address_space(3))) v4i_vs* lds_v4i_p;

#if defined(__AMDGCN__) &&                                                   \
    __has_builtin(__builtin_amdgcn_global_load_async_to_lds_b128) &&         \
    __has_builtin(__builtin_amdgcn_s_wait_asynccnt)
#define FS_USE_ASYNC 1
#else
#define FS_USE_ASYNC 0
#endif

// h/d/T are fixed literal constants in the reference -> bake as immediates.
__device__ __forceinline__ float fs16(float x) {
  constexpr float H[16] = {0.4462f, 0.9426f, 0.5828f, 0.2679f,
                           0.1929f, 1.1032f, 0.0062f, 1.7608f,
                           1.6892f, 1.0465f, 2.2203f, -0.0518f,
                           0.9965f, 1.2357f, 0.7535f, 1.3039f};
  constexpr float D[16] = {0.1441f, 1.0263f, 0.5819f, 0.2583f,
                           0.089f,  0.8074f, 0.1049f, 1.2033f,
                           1.8082f, 0.4312f, 2.2586f, -0.2693f,
                           0.8391f, 0.0463f, 0.2339f, 0.1115f};
  constexpr float T[16] = {-0.4326f, 0.7987f, 0.1965f, -0.0293f,
                           1.7898f,  0.4043f, -0.1738f, -0.0356f,
                           2.1835f, -0.0467f, 2.3067f, -1.7284f,
                           1.281f,   0.942f, -0.245f,  -0.5279f};
  float v = x;
  float o = 0.0f;
#pragma unroll
  for (int t = 0; t < 16; ++t) {
    // (v - T > 0) == (v > T) for finite IEEE floats with denorms preserved.
    float zf = (v > T[t]) ? 1.0f : 0.0f;       // v_cmp + v_cndmask
    v = __builtin_fmaf(zf, -H[t], v);          // exact: v or v-H
    o = __builtin_fmaf(zf, D[t], o);           // exact: o or o+D
  }
  return o;
}

__device__ __forceinline__ v4f fs16x4(v4f in) {
  // Four independent serial chains -> scheduler interleaves / dual-issues.
  v4f r;
  r.x = fs16(in.x);
  r.y = fs16(in.y);
  r.z = fs16(in.z);
  r.w = fs16(in.w);
  return r;
}

#if FS_USE_ASYNC
__device__ __forceinline__ void async_copy16(const v4f* gsrc, v4f* ldst) {
  // GLOBAL_LOAD_ASYNC_TO_LDS_B128: per-lane 16B global -> LDS, ASYNCcnt.
  v4i_vs* g0 = (v4i_vs*)(void*)(v4f*)gsrc;  // strip const, retype (generic AS)
  v4i_vs* l0 = (v4i_vs*)(void*)ldst;
  __builtin_amdgcn_global_load_async_to_lds_b128(
      (gbl_v4i_p)g0,   // addrspacecast generic -> global, same pointee
      (lds_v4i_p)l0,   // addrspacecast generic -> local,  same pointee
      /*offset=*/0, /*cpol=*/0);
}
#endif

__global__ __launch_bounds__(256) void fs_swish_kernel(
    const float* __restrict__ x, float* __restrict__ out,
    long long n, long long trips) {
  const long long n4   = n >> 2;
  const long long tid  = (long long)blockIdx.x * blockDim.x + threadIdx.x;
  const long long nthr = (long long)gridDim.x * blockDim.x;

  const v4f* __restrict__ x4 = (const v4f*)x;
  v4f* __restrict__ out4     = (v4f*)out;

#if FS_USE_ASYNC
  // 4-deep async ring: 4 stages x 256 lanes x 16B = 16 KB LDS.
  __shared__ v4f buf[4][256];
  if (n4 > 0) {
    const long long nm1 = n4 - 1;
    // Clamped index: every lane ALWAYS issues (wave-uniform ASYNCcnt math).
#define FS_CLAMP_IDX(m) ((tid + (m) * nthr) < n4 ? (tid + (m) * nthr) : nm1)
    // Prologue: stages 0..2 in flight.
#pragma unroll
    for (int p = 0; p < 3; ++p)
      async_copy16(x4 + FS_CLAMP_IDX(p), &buf[p][threadIdx.x]);

    for (long long k = 0; k < trips; ++k) {
      // Issue stage k+3, then retire stage k (async loads complete in order).
      async_copy16(x4 + FS_CLAMP_IDX(k + 3), &buf[(k + 3) & 3][threadIdx.x]);
      __builtin_amdgcn_s_wait_asynccnt(3);
      __asm__ __volatile__("" ::: "memory");  // no LDS reads above the wait

      v4f val = buf[k & 3][threadIdx.x];      // ds_load_b128
      v4f res = fs16x4(val);

      long long idx = tid + k * nthr;
      if (idx < n4) __builtin_nontemporal_store(res, out4 + idx);
    }
#undef FS_CLAMP_IDX
  }
#else
  for (long long k = 0; k < trips; ++k) {
    long long idx = tid + k * nthr;
    if (idx < n4) {
      v4f val = __builtin_nontemporal_load(x4 + idx);
      __builtin_nontemporal_store(fs16x4(val), out4 + idx);
    }
  }
#endif

  // Scalar tail for n % 4 (dead for the reference shape, kept for safety).
  for (long long i = (n4 << 2) + tid; i < n; i += nthr) {
    out[i] = fs16(x[i]);
  }
}

extern "C" void kernel_launch(void* const* d_in, const int* in_sizes, int n_in,
                              void* d_out, int out_size, void* d_ws,
                              size_t ws_size, hipStream_t stream) {
  const float* x = (const float*)d_in[0];
  float* out     = (float*)d_out;
  long long n    = (long long)in_sizes[0];  // 64*256*64*64 = 67,108,864

  const int threads   = 256;               // 8 wave32s per block
  long long n4        = n >> 2;
  long long wantBlks  = (n4 + threads - 1) / threads;
  if (wantBlks < 1) wantBlks = 1;
  int blocks = (int)((wantBlks < 8192) ? wantBlks : 8192);

  // Uniform per-thread trip count, computed on host (no device 64-bit div).
  long long nthr  = (long long)blocks * threads;
  long long trips = (n4 + nthr - 1) / nthr;

  fs_swish_kernel<<<blocks, threads, 0, stream>>>(x, out, n, trips);
}